// GCNEncoder_75127567941896
// MI455X (gfx1250) — compile-verified
//
#include <hip/hip_runtime.h>
#include <hip/hip_bf16.h>

// ---------------------------------------------------------------------------
// GCN 2-layer forward for gfx1250 (MI455X).
//   layer: h = A_norm @ (x @ W) + b   (relu after layer 1)
// Plan:
//   1) degree count (int atomics) -> dinv = rsqrt(deg+1)
//   2) single-block exclusive scan -> rowptr  (self-loop reserved per node)
//   3) CSR fill: packed 8B records (col:int, w:float), w = dinv[r]*dinv[c]
//   4) GEMM via V_WMMA_F32_16X16X4_F32 (fp32-exact matrix path)
//   5) node-parallel aggregation: one wave32 per node, float2 per lane,
//      4-wide batched neighbor gathers (coalesced 256B bursts) from the
//      L2-resident feature table — zero float atomics.
// ---------------------------------------------------------------------------

typedef float v2f __attribute__((ext_vector_type(2)));
typedef float v8f __attribute__((ext_vector_type(8)));
typedef int   v2i __attribute__((ext_vector_type(2)));

__global__ void init_cnt_kernel(int* __restrict__ cnt, int N) {
    int i = blockIdx.x * blockDim.x + threadIdx.x;
    if (i < N) cnt[i] = 0;
}

__global__ void count_deg_kernel(const int* __restrict__ ei, int* __restrict__ cnt, int E) {
    int e = blockIdx.x * blockDim.x + threadIdx.x;
    if (e < E) atomicAdd(&cnt[ei[e]], 1);   // ei[0..E) = row (targets)
}

__global__ void dinv_kernel(const int* __restrict__ cnt, float* __restrict__ dinv, int N) {
    int i = blockIdx.x * blockDim.x + threadIdx.x;
    if (i < N) dinv[i] = rsqrtf((float)(cnt[i] + 1));   // +1 self-loop; deg >= 1 always
}

// Single-block exclusive scan of (cnt[i]+1) over N elements -> rowptr[0..N].
__global__ __launch_bounds__(1024) void scan_kernel(const int* __restrict__ cnt,
                                                    int* __restrict__ rowptr, int N) {
    __shared__ int sdata[1024];
    int t = threadIdx.x;
    int chunk = (N + 1023) / 1024;
    int begin = t * chunk;
    int end   = begin + chunk;
    if (begin > N) begin = N;
    if (end   > N) end   = N;
    int sum = 0;
    for (int i = begin; i < end; ++i) sum += cnt[i] + 1;
    sdata[t] = sum;
    __syncthreads();
    // Hillis-Steele inclusive scan over 1024 partials
    for (int off = 1; off < 1024; off <<= 1) {
        int v = (t >= off) ? sdata[t - off] : 0;
        __syncthreads();
        sdata[t] += v;
        __syncthreads();
    }
    int run = (t == 0) ? 0 : sdata[t - 1];  // exclusive prefix of this chunk
    for (int i = begin; i < end; ++i) {
        rowptr[i] = run;
        run += cnt[i] + 1;
    }
    if (t == 1023) rowptr[N] = sdata[1023]; // total = N + E
}

__global__ void init_csr_kernel(const int* __restrict__ rowptr, const float* __restrict__ dinv,
                                int* __restrict__ cursor, v2i* __restrict__ csr,
                                int N) {
    int i = blockIdx.x * blockDim.x + threadIdx.x;
    if (i < N) {
        int p = rowptr[i];
        cursor[i] = p + 1;           // slot 0 of each segment = self-loop
        float d = dinv[i];
        v2i rec;
        rec.x = i;
        rec.y = __float_as_int(d * d);
        csr[p] = rec;                // single b64 store
    }
}

__global__ void fill_csr_kernel(const int* __restrict__ ei, const float* __restrict__ dinv,
                                int* __restrict__ cursor, v2i* __restrict__ csr, int E) {
    int e = blockIdx.x * blockDim.x + threadIdx.x;
    if (e < E) {
        int r = ei[e];        // target
        int c = ei[E + e];    // source
        int p = atomicAdd(&cursor[r], 1);
        v2i rec;
        rec.x = c;
        rec.y = __float_as_int(dinv[r] * dinv[c]);
        csr[p] = rec;         // single b64 store
    }
}

// C[M,64] = A[M,K] @ B[K,64], fp32 WMMA. 256 threads = 8 waves; each wave owns a
// 16x16 C tile; block covers 32 rows x 64 cols. Requires M % 32 == 0 for full
// tiles (true here: 100000 = 3125*32), K % 4 == 0.
__global__ __launch_bounds__(256) void gemm64_wmma_kernel(const float* __restrict__ A,
                                                          const float* __restrict__ B,
                                                          float* __restrict__ C,
                                                          int M, int K) {
    int tid  = threadIdx.x;
    int wave = tid >> 5;          // 0..7
    int lane = tid & 31;
    int m0 = blockIdx.x * 32 + (wave >> 2) * 16;
    int n0 = (wave & 3) * 16;
    if (m0 >= M) return;          // wave-uniform; EXEC stays all-ones for WMMA
    int half = lane >> 4;         // 0: K pair {k,k+1}, 1: {k+2,k+3}
    int l    = lane & 15;

    const float* arow = A + (size_t)(m0 + l) * K;
    v8f acc = {0.f, 0.f, 0.f, 0.f, 0.f, 0.f, 0.f, 0.f};

    for (int k = 0; k < K; k += 4) {
        int ka = k + 2 * half;
        // A 16x4 tile: lanes 0-15 hold K={k,k+1}, lanes 16-31 hold K={k+2,k+3}
        v2f a = *(const v2f*)(arow + ka);
        // B 4x16 tile: VGPR0 = row ka, VGPR1 = row ka+1, N striped over lanes
        v2f b;
        b.x = B[(size_t)ka * 64 + n0 + l];
        b.y = B[(size_t)(ka + 1) * 64 + n0 + l];
        acc = __builtin_amdgcn_wmma_f32_16x16x4_f32(false, a, false, b,
                                                    (short)0, acc, false, false);
    }
    // C layout: VGPR j -> row m0 + j + 8*half, col n0 + l
    float* crow = C + (size_t)m0 * 64 + n0 + l;
#pragma unroll
    for (int j = 0; j < 8; ++j) {
        crow[(size_t)(j + 8 * half) * 64] = acc[j];
    }
}

// One wave32 per node; lane owns features {2*lane, 2*lane+1}. Neighbor gathers
// are coalesced 32x8B = 256B bursts from the L2-resident feature table.
// Batched 4-wide: 4 edge records loaded together, then 4 independent feature
// gathers in flight before the FMAs (hides L2 latency with MLP).
__global__ __launch_bounds__(256) void agg_kernel(const float* __restrict__ h,
                                                  const int* __restrict__ rowptr,
                                                  const v2i* __restrict__ csr,
                                                  const float* __restrict__ bias,
                                                  float* __restrict__ out,
                                                  int N, int do_relu) {
    int node = blockIdx.x * 8 + (threadIdx.x >> 5);
    if (node >= N) return;
    int lane = threadIdx.x & 31;

    const float* hbase = h + 2 * lane;

    float ax = 0.f, ay = 0.f;
    int j = rowptr[node];
    int e = rowptr[node + 1];

    for (; j + 3 < e; j += 4) {
        v2i e0 = csr[j];
        v2i e1 = csr[j + 1];
        v2i e2 = csr[j + 2];
        v2i e3 = csr[j + 3];
        v2f h0 = *(const v2f*)(hbase + (size_t)e0.x * 64);
        v2f h1 = *(const v2f*)(hbase + (size_t)e1.x * 64);
        v2f h2 = *(const v2f*)(hbase + (size_t)e2.x * 64);
        v2f h3 = *(const v2f*)(hbase + (size_t)e3.x * 64);
        float w0 = __int_as_float(e0.y);
        float w1 = __int_as_float(e1.y);
        float w2 = __int_as_float(e2.y);
        float w3 = __int_as_float(e3.y);
        ax = fmaf(h0.x, w0, ax); ay = fmaf(h0.y, w0, ay);
        ax = fmaf(h1.x, w1, ax); ay = fmaf(h1.y, w1, ay);
        ax = fmaf(h2.x, w2, ax); ay = fmaf(h2.y, w2, ay);
        ax = fmaf(h3.x, w3, ax); ay = fmaf(h3.y, w3, ay);
    }
    for (; j < e; ++j) {
        v2i er = csr[j];
        float w = __int_as_float(er.y);
        v2f hv = *(const v2f*)(hbase + (size_t)er.x * 64);
        ax = fmaf(hv.x, w, ax);
        ay = fmaf(hv.y, w, ay);
    }

    v2f bv = *(const v2f*)(bias + 2 * lane);
    ax += bv.x;
    ay += bv.y;
    if (do_relu) {
        ax = fmaxf(ax, 0.f);
        ay = fmaxf(ay, 0.f);
    }
    v2f o; o.x = ax; o.y = ay;
    *(v2f*)(out + (size_t)node * 64 + 2 * lane) = o;
}

extern "C" void kernel_launch(void* const* d_in, const int* in_sizes, int n_in,
                              void* d_out, int out_size, void* d_ws, size_t ws_size,
                              hipStream_t stream) {
    (void)n_in; (void)out_size; (void)ws_size;

    const float* x  = (const float*)d_in[0];
    const int*   ei = (const int*)d_in[1];
    const float* W1 = (const float*)d_in[2];
    const float* b1 = (const float*)d_in[3];
    const float* W2 = (const float*)d_in[4];
    const float* b2 = (const float*)d_in[5];
    float* out = (float*)d_out;

    const int H   = in_sizes[3];            // 64
    const int IN  = in_sizes[2] / H;        // 128
    const int N   = in_sizes[0] / IN;       // 100000
    const int E   = in_sizes[1] / 2;        // 1600000
    const int NNZ = N + E;                  // edges + self-loops

    // bump allocator over workspace, 256B aligned
    char* ws = (char*)d_ws;
    size_t off = 0;
    auto alloc = [&](size_t bytes) -> void* {
        void* p = (void*)(ws + off);
        off += (bytes + 255) & ~(size_t)255;
        return p;
    };
    int*   cnt     = (int*)  alloc((size_t)N * 4);
    int*   rowptr  = (int*)  alloc((size_t)(N + 1) * 4);
    int*   cursor  = (int*)  alloc((size_t)N * 4);
    float* dinv    = (float*)alloc((size_t)N * 4);
    v2i*   csr     = (v2i*)  alloc((size_t)NNZ * 8);   // packed (col, weight)
    float* h1      = (float*)alloc((size_t)N * H * 4);
    float* y1      = (float*)alloc((size_t)N * H * 4);
    float* h2      = h1;   // reuse: h1 dead after first aggregation

    const int T = 256;
    int gN = (N + T - 1) / T;
    int gE = (E + T - 1) / T;
    int gAgg = (N + 7) / 8;         // one wave per node, 8 waves per block
    int gGemm = (N + 31) / 32;      // 32 rows per block

    init_cnt_kernel<<<gN, T, 0, stream>>>(cnt, N);
    count_deg_kernel<<<gE, T, 0, stream>>>(ei, cnt, E);
    dinv_kernel<<<gN, T, 0, stream>>>(cnt, dinv, N);
    scan_kernel<<<1, 1024, 0, stream>>>(cnt, rowptr, N);
    init_csr_kernel<<<gN, T, 0, stream>>>(rowptr, dinv, cursor, csr, N);
    fill_csr_kernel<<<gE, T, 0, stream>>>(ei, dinv, cursor, csr, E);

    // Layer 1: h1 = x @ W1 ; y1 = relu(A_norm @ h1 + b1)
    gemm64_wmma_kernel<<<gGemm, T, 0, stream>>>(x, W1, h1, N, IN);
    agg_kernel<<<gAgg, T, 0, stream>>>(h1, rowptr, csr, b1, y1, N, 1);

    // Layer 2: h2 = y1 @ W2 ; out = A_norm @ h2 + b2
    gemm64_wmma_kernel<<<gGemm, T, 0, stream>>>(y1, W2, h2, N, H);
    agg_kernel<<<gAgg, T, 0, stream>>>(h2, rowptr, csr, b2, out, N, 0);
}